// PointPillarsScatter_43619687858551
// MI455X (gfx1250) — compile-verified
//
#include <hip/hip_runtime.h>
#include <stdint.h>

// PointPillars BEV scatter for MI455X (gfx1250, wave32).
// out[b, c, y, x] += feats[b, p, c] for pillar p at (y, x).
// B=4, P=30000, C=64, H=W=512 -> 268 MB f32 output, pure memory-bound scatter.
//
// Phase 1: zero-fill 268 MB via Tensor Data Mover (TENSOR_STORE_FROM_LDS with
//          iterate_enable: one descriptor DMAs 4 MB of zeros from a 64 KB LDS
//          region; 64 workgroups cover the whole output).
// Phase 2: scatter-add with native no-return global_atomic_add_f32.

typedef float        v4f   __attribute__((ext_vector_type(4)));
typedef int          v2i   __attribute__((ext_vector_type(2)));
typedef unsigned int u32x4 __attribute__((ext_vector_type(4)));
typedef int          i32x4 __attribute__((ext_vector_type(4)));
typedef int          i32x8 __attribute__((ext_vector_type(8)));

#define BEV_H   512
#define BEV_W   512
#define BEV_HW  (BEV_H * BEV_W)
#define C_CH    64

#define TDM_TILE_ELEMS 16384   // 64 KB tile per TDM iteration (== LDS region)
#define TDM_ITERS      64      // iterations per descriptor -> 4 MB per block

// ---------------------------------------------------------------------------
// Zero-fill via TDM: zero 64 KB of LDS once, then a single
// TENSOR_STORE_FROM_LDS descriptor (iterate_enable=1, lds_addr_increment=0,
// global_addr_increment=64 KB) streams 64 copies of it to global memory.
// ---------------------------------------------------------------------------
__global__ void bev_zero_tdm_kernel(float* __restrict__ out) {
    __shared__ v4f lds4[TDM_TILE_ELEMS / 4];   // 64 KB

    // Cooperative LDS zero (B128 DS stores).
    for (int i = threadIdx.x; i < TDM_TILE_ELEMS / 4; i += blockDim.x)
        lds4[i] = (v4f)0.0f;
    __syncthreads();

    if (threadIdx.x < 32) {  // wave 0 issues the DMA descriptor
        // Ensure our DS stores are visible to the TDM engine.
        asm volatile("s_wait_dscnt 0" ::: "memory");

        const uint64_t base =
            (uint64_t)(uintptr_t)out +
            (uint64_t)blockIdx.x * ((uint64_t)TDM_TILE_ELEMS * TDM_ITERS * 4u);
        // Flat LDS address low 32 bits == wave-relative LDS byte offset.
        const uint32_t lds_off = (uint32_t)(uintptr_t)&lds4[0];

        // D# group 0: count=1, lds_addr, global_addr[56:0], type=2.
        u32x4 g0;
        g0.x = 1u;
        g0.y = lds_off;
        g0.z = (uint32_t)base;
        g0.w = (uint32_t)((base >> 32) & 0x01FFFFFFu) | (2u << 30);

        // D# group 1: data_size=4B (code 2), iterate_enable=1,
        // tensor_dim0 = tile_dim0 = 16384, tensor_dim1 = tile_dim1 = 1,
        // tensor_dim0_stride = 16384.
        i32x8 g1;
        g1[0] = (2 << 16) | (1 << 19);                     // data_size | iterate
        g1[1] = (int)((TDM_TILE_ELEMS & 0xFFFF) << 16);    // tensor_dim0 lo16
        g1[2] = (TDM_TILE_ELEMS >> 16) | (1 << 16);        // dim0 hi16 | tensor_dim1=1
        g1[3] = (int)((TDM_TILE_ELEMS & 0xFFFF) << 16);    // dim1 hi=0 | tile_dim0
        g1[4] = 1;                                         // tile_dim1=1, tile_dim2=0
        g1[5] = TDM_TILE_ELEMS;                            // dim0_stride lo32
        g1[6] = 0;                                         // stride hi / dim1_stride
        g1[7] = 0;

        // D# group 2 (iterate mode): lds_addr_inc=0 (reuse zeros),
        // global_addr_inc = 16384 elements (64 KB), iterate_count = ITERS-1.
        i32x4 g2;
        g2[0] = 1;                                         // tensor_dim2
        g2[1] = 0;                                         // lds_addr_increment
        g2[2] = TDM_TILE_ELEMS;                            // global_addr_increment
        g2[3] = (TDM_ITERS - 1) << 16;                     // iterate_count

        i32x4 g3 = {0, 0, 0, 0};                           // unused (2D + iterate)

        // 6-arg toolchain form: extra (unused VADDR4) int32x8 group, zeroed.
        i32x8 g4 = {0, 0, 0, 0, 0, 0, 0, 0};

        __builtin_amdgcn_tensor_store_from_lds(g0, g1, g2, g3, g4, 0);
        __builtin_amdgcn_s_wait_tensorcnt(0);
    }
}

// ---------------------------------------------------------------------------
// Fallback zero-fill (plain B128 stores); not taken for the fixed shapes.
// ---------------------------------------------------------------------------
__global__ void bev_zero_kernel(float* __restrict__ out, long long n4) {
    long long i      = (long long)blockIdx.x * blockDim.x + threadIdx.x;
    long long stride = (long long)gridDim.x * blockDim.x;
    v4f z = (v4f)0.0f;
    for (; i < n4; i += stride)
        ((v4f*)out)[i] = z;
}

// ---------------------------------------------------------------------------
// Scatter-add: 16 threads per pillar, NT B128 feature loads (read-once data,
// preserve L2 for the output), 4 no-return f32 global atomics per thread into
// the channel planes (1 MB apart -> absorbed by the L2 atomic units).
// ---------------------------------------------------------------------------
__global__ void bev_scatter_kernel(const float* __restrict__ feats,
                                   const int*   __restrict__ coords,
                                   float*       __restrict__ out,
                                   int P) {
    const int b   = blockIdx.y;
    const int tid = blockIdx.x * blockDim.x + threadIdx.x;
    const int p   = tid >> 4;
    if (p >= P) return;
    const int c4  = (tid & 15) << 2;

    const long long pil = (long long)b * P + p;

    const v2i yx = *(const v2i*)(coords + (pil << 1));
    int y = yx.x; int x = yx.y;
    y = min(max(y, 0), BEV_H - 1);
    x = min(max(x, 0), BEV_W - 1);
    const int flat = (y << 9) | x;

    const v4f f = __builtin_nontemporal_load(
        (const v4f*)(feats + (pil << 6) + c4));

    float* o = out + ((long long)(b * C_CH + c4)) * BEV_HW + flat;

    unsafeAtomicAdd(o,              f.x);
    unsafeAtomicAdd(o +     BEV_HW, f.y);
    unsafeAtomicAdd(o + 2 * BEV_HW, f.z);
    unsafeAtomicAdd(o + 3 * BEV_HW, f.w);
}

// ---------------------------------------------------------------------------
// Launch: TDM zero-fill then scatter, in order on `stream`.
// ---------------------------------------------------------------------------
extern "C" void kernel_launch(void* const* d_in, const int* in_sizes, int n_in,
                              void* d_out, int out_size, void* d_ws, size_t ws_size,
                              hipStream_t stream) {
    const float* feats  = (const float*)d_in[0];   // [B, P, C] f32
    const int*   coords = (const int*)  d_in[1];   // [B, P, 2] i32
    float*       out    = (float*)d_out;           // [B, C, H, W] f32

    const int B = 4;
    const int P = in_sizes[1] / (B * 2);           // 30000

    const long long elems     = (long long)out_size;
    const long long per_block = (long long)TDM_TILE_ELEMS * TDM_ITERS;

    if (elems % per_block == 0) {
        const int blocks = (int)(elems / per_block);   // 64 for these shapes
        bev_zero_tdm_kernel<<<blocks, 256, 0, stream>>>(out);
    } else {
        bev_zero_kernel<<<2048, 256, 0, stream>>>(out, elems / 4);
    }

    const int threads_per_batch = P * 16;
    dim3 grid((threads_per_batch + 255) / 256, B);
    bev_scatter_kernel<<<grid, 256, 0, stream>>>(feats, coords, out, P);
}